// LengthRegulator_88304527606014
// MI455X (gfx1250) — compile-verified
//
#include <hip/hip_runtime.h>

// Problem constants (from reference): B=16, N=512, C=512, L=2048.
// out[b,l,c] = enc[b, seg(b,l), c]  where seg is the duration segment containing l, else 0.

#define LR_B 16
#define LR_N 512
#define LR_C 512
#define LR_L 2048
#define WPB  8   // wave32 rows (waves) per block in the gather kernel

#define HAS_ASYNC_LDS ( __has_builtin(__builtin_amdgcn_global_load_async_to_lds_b128) && \
                        __has_builtin(__builtin_amdgcn_global_store_async_from_lds_b128) && \
                        __has_builtin(__builtin_amdgcn_s_wait_asynccnt) )

#if HAS_ASYNC_LDS
// Builtin signature (from hipcc diagnostic): param0/1 are pointers to a
// 16-byte int vector, in global (AS1) and LDS (AS3) address spaces.
typedef int v4i __attribute__((vector_size(16)));
typedef __attribute__((address_space(1))) v4i* gv4i_p;   // global
typedef __attribute__((address_space(3))) v4i* lv4i_p;   // LDS
#endif

// ---------------------------------------------------------------------------
// Kernel 1: build per-timestep segment index map idx[b][l] in d_ws.
// One block per batch, 512 threads (= N). Inclusive scan of durations in LDS,
// then scatter thread index n into idx[start..end).
// ---------------------------------------------------------------------------
__global__ __launch_bounds__(LR_N) void lr_index_kernel(const float* __restrict__ logd,
                                                        int* __restrict__ idx) {
    __shared__ int s_scan[LR_N];
    const int b = blockIdx.x;
    const int n = threadIdx.x;

    const float ld = logd[b * LR_N + n];
    int dur = 0;
    if (ld > 0.0f) dur = (int)floorf(exp2f(ld) + 1e-4f);

    s_scan[n] = dur;
    __syncthreads();
    // Hillis-Steele inclusive scan over 512 elements.
    #pragma unroll
    for (int off = 1; off < LR_N; off <<= 1) {
        const int add = (n >= off) ? s_scan[n - off] : 0;
        __syncthreads();
        s_scan[n] += add;
        __syncthreads();
    }
    const int end   = s_scan[n];
    const int start = end - dur;

    int* __restrict__ row = idx + b * LR_L;
    // init whole map to -1 (rows past total duration stay zero-filled)
    #pragma unroll
    for (int l = n; l < LR_L; l += LR_N) row[l] = -1;
    __syncthreads();
    const int e = end < LR_L ? end : LR_L;
    for (int l = start; l < e; ++l) row[l] = n;
}

// ---------------------------------------------------------------------------
// Kernel 2: gather encoder rows into output. One wave32 per output row.
// Row copy = 512 f32 = 2048 B = 4 x (32 lanes x 16 B) async DMA transfers.
// ---------------------------------------------------------------------------
__global__ __launch_bounds__(WPB * 32) void lr_gather_kernel(const float* __restrict__ enc,
                                                             const int* __restrict__ idx,
                                                             float* __restrict__ out) {
#if HAS_ASYNC_LDS
    __shared__ __align__(16) float lds[WPB * LR_C];   // 16 KB
#endif
    const int wave = threadIdx.x >> 5;
    const int lane = threadIdx.x & 31;
    const int row  = blockIdx.x * WPB + wave;         // [0, B*L)
    const int b    = row >> 11;                       // row / L (L = 2048)
    const int n    = idx[row];                        // wave-uniform

    float* __restrict__ orow = out + (size_t)row * LR_C;

    if (n >= 0) {
        const float* __restrict__ src = enc + ((size_t)b * LR_N + n) * LR_C;
#if HAS_ASYNC_LDS
        float* wlds = lds + wave * LR_C;
        // global -> LDS, 512 B per instruction across the wave
        #pragma unroll
        for (int i = 0; i < 4; ++i) {
            __builtin_amdgcn_global_load_async_to_lds_b128(
                (gv4i_p)(src + i * 128 + lane * 4),
                (lv4i_p)(wlds + i * 128 + lane * 4),
                0, 0);
        }
        __builtin_amdgcn_s_wait_asynccnt(0);
        // LDS -> global into the output row
        #pragma unroll
        for (int i = 0; i < 4; ++i) {
            __builtin_amdgcn_global_store_async_from_lds_b128(
                (gv4i_p)(orow + i * 128 + lane * 4),
                (lv4i_p)(wlds + i * 128 + lane * 4),
                0, 0);
        }
        // implicit wait-idle at s_endpgm drains remaining async stores
#else
        const float4* __restrict__ s4 = (const float4*)src;
        float4* __restrict__ o4 = (float4*)orow;
        #pragma unroll
        for (int i = 0; i < 4; ++i) o4[lane + i * 32] = s4[lane + i * 32];
#endif
    } else {
        const float4 z = make_float4(0.f, 0.f, 0.f, 0.f);
        float4* __restrict__ o4 = (float4*)orow;
        #pragma unroll
        for (int i = 0; i < 4; ++i) o4[lane + i * 32] = z;
    }
}

// ---------------------------------------------------------------------------
extern "C" void kernel_launch(void* const* d_in, const int* in_sizes, int n_in,
                              void* d_out, int out_size, void* d_ws, size_t ws_size,
                              hipStream_t stream) {
    const float* enc  = (const float*)d_in[0];   // (B, N, C) f32
    const float* logd = (const float*)d_in[1];   // (B, N, 1) f32
    float*       out  = (float*)d_out;           // (B, L, C) f32
    int*         idx  = (int*)d_ws;              // B*L ints = 128 KB scratch

    lr_index_kernel<<<LR_B, LR_N, 0, stream>>>(logd, idx);

    const int total_rows = LR_B * LR_L;          // 32768 rows
    lr_gather_kernel<<<total_rows / WPB, WPB * 32, 0, stream>>>(enc, idx, out);
}